// DynamicGNN_66958540145299
// MI455X (gfx1250) — compile-verified
//
#include <hip/hip_runtime.h>
#include <math.h>

// ---------------------------------------------------------------------------
// GraphSAGE (3 hidden SAGEConv+ReLU+BN layers, D=128) + SAGEConv(D->1)+sigmoid
// fp32 end-to-end; GEMMs run on V_WMMA_F32_16X16X4_F32 (exact fp32 tensor op).
// ---------------------------------------------------------------------------

#define D       128
#define LAYERS  3
#define BN_EPS  1e-5f

typedef float v2f __attribute__((ext_vector_type(2)));
typedef float v8f __attribute__((ext_vector_type(8)));

// -------------------------------- utility ----------------------------------

__global__ void fill_zero_kernel(float* __restrict__ p, int n) {
  int i = blockIdx.x * blockDim.x + threadIdx.x;
  if (i < n) p[i] = 0.0f;
}

__global__ void deg_kernel(const int* __restrict__ dst, float* __restrict__ deg,
                           int nE, int nN) {
  int e = blockIdx.x * blockDim.x + threadIdx.x;
  if (e >= nE) return;
  int d = dst[e];
  if ((unsigned)d < (unsigned)nN) atomicAdd(&deg[d], 1.0f);
}

__global__ void invdeg_kernel(const float* __restrict__ deg,
                              float* __restrict__ inv, int nN) {
  int i = blockIdx.x * blockDim.x + threadIdx.x;
  if (i >= nN) return;
  float d = deg[i];
  inv[i] = (d > 0.0f) ? (1.0f / d) : 0.0f;
}

// -------------------------- edge scatter (sum) -----------------------------
// One wave32 per edge; 32 lanes sweep the 128 features in 4 coalesced bursts.
// agg[dst] += h[src]   (mean applied later via inv_deg during GEMM staging)

__global__ __launch_bounds__(256)
void scatter_kernel(const int* __restrict__ src, const int* __restrict__ dst,
                    const float* __restrict__ h, float* __restrict__ agg,
                    int nE, int nN) {
  int tid  = blockIdx.x * blockDim.x + threadIdx.x;
  int e    = tid >> 5;           // wave32 per edge
  int lane = tid & 31;
  if (e >= nE) return;
  int s = src[e], d = dst[e];
  if ((unsigned)s >= (unsigned)nN || (unsigned)d >= (unsigned)nN) return;
  const float* hp = h   + (size_t)s * D;
  float*       ap = agg + (size_t)d * D;
#pragma unroll
  for (int c = 0; c < 4; ++c) {
    int f = lane + 32 * c;
    atomicAdd(&ap[f], hp[f]);
  }
}

// --------------------- fused SAGE GEMM + bias + ReLU -----------------------
// out[16 x 128 strip] = relu( (agg*inv_deg) @ Wl + h @ Wr + b )
// Block = 256 threads = 8 waves; each wave owns one 16-col tile.
// A strips staged in LDS (inv_deg folded in); WMMA fp32 16x16x4, K swept 0..127.

__global__ __launch_bounds__(256)
void sage_gemm_relu_kernel(const float* __restrict__ agg,
                           const float* __restrict__ h,
                           const float* __restrict__ invdeg,
                           const float* __restrict__ Wl,
                           const float* __restrict__ Wr,
                           const float* __restrict__ bias,
                           float* __restrict__ out, int nN) {
  __shared__ float sA[16][D];   // mean-aggregated rows (pre-scaled)
  __shared__ float sH[16][D];   // root rows
  const int row0 = blockIdx.x * 16;
  const int tid  = threadIdx.x;

  // Stage 16x128 of agg (scaled by inv_deg) and h: 8 floats per thread each.
  {
    int base = tid * 8;
    int r = base >> 7;          // 0..15
    int c = base & 127;         // multiple of 8
    int row = row0 + r;
    if (row < nN) {
      float sc = invdeg[row];
      const float4* ga = (const float4*)(agg + (size_t)row * D + c);
      const float4* gh = (const float4*)(h   + (size_t)row * D + c);
      float4 a0 = ga[0], a1 = ga[1];
      float4 h0 = gh[0], h1 = gh[1];
      float4* pa = (float4*)&sA[r][c];
      float4* ph = (float4*)&sH[r][c];
      pa[0] = make_float4(a0.x * sc, a0.y * sc, a0.z * sc, a0.w * sc);
      pa[1] = make_float4(a1.x * sc, a1.y * sc, a1.z * sc, a1.w * sc);
      ph[0] = h0; ph[1] = h1;
    } else {
      float4 z = make_float4(0.f, 0.f, 0.f, 0.f);
      ((float4*)&sA[r][c])[0] = z; ((float4*)&sA[r][c])[1] = z;
      ((float4*)&sH[r][c])[0] = z; ((float4*)&sH[r][c])[1] = z;
    }
  }
  __syncthreads();

  const int wave = tid >> 5;          // 0..7 -> output col tile
  const int lane = tid & 31;
  const int n0   = wave * 16;
  const int arow = lane & 15;         // A fragment row (M)
  const int kb   = (lane >> 4) * 2;   // K sub-offset: 0 (lanes<16) or 2
  const int ncol = lane & 15;         // B fragment col (N)

  v8f acc = {};
#if __has_builtin(__builtin_amdgcn_wmma_f32_16x16x4_f32)
#pragma unroll
  for (int k0 = 0; k0 < D; k0 += 4) {
    int k = k0 + kb;
    v2f a, b;
    // neighbor term: (agg*inv_deg) @ Wl
    a.x = sA[arow][k];  a.y = sA[arow][k + 1];
    b.x = Wl[(size_t)k       * D + n0 + ncol];
    b.y = Wl[(size_t)(k + 1) * D + n0 + ncol];
    acc = __builtin_amdgcn_wmma_f32_16x16x4_f32(false, a, false, b,
                                                (short)0, acc, false, false);
    // root term: h @ Wr
    a.x = sH[arow][k];  a.y = sH[arow][k + 1];
    b.x = Wr[(size_t)k       * D + n0 + ncol];
    b.y = Wr[(size_t)(k + 1) * D + n0 + ncol];
    acc = __builtin_amdgcn_wmma_f32_16x16x4_f32(false, a, false, b,
                                                (short)0, acc, false, false);
  }
#else
  // Scalar fallback (keeps everything else compiling; histogram shows wmma=0)
  {
    const int mbase = (lane < 16) ? 0 : 8;
    const int n = n0 + ncol;
#pragma unroll
    for (int j = 0; j < 8; ++j) {
      float s = 0.f;
      for (int k = 0; k < D; ++k)
        s += sA[mbase + j][k] * Wl[(size_t)k * D + n] +
             sH[mbase + j][k] * Wr[(size_t)k * D + n];
      acc[j] = s;
    }
  }
#endif

  // Epilogue: C layout -> VGPR j: lanes 0-15 hold M=j, lanes 16-31 hold M=j+8
  const int n     = n0 + ncol;
  const float bv  = bias[n];
  const int mbase = (lane < 16) ? 0 : 8;
#pragma unroll
  for (int j = 0; j < 8; ++j) {
    int m = row0 + mbase + j;
    if (m < nN) {
      float v = acc[j] + bv;
      out[(size_t)m * D + n] = (v > 0.0f) ? v : 0.0f;
    }
  }
}

// ------------------------------ batchnorm ----------------------------------

__global__ __launch_bounds__(128)
void bn_stats_kernel(const float* __restrict__ h, float* __restrict__ sum,
                     float* __restrict__ sq, int nN) {
  const int col = threadIdx.x;           // blockDim == 128
  const int r0  = blockIdx.x * 256;
  int rend = r0 + 256; if (rend > nN) rend = nN;
  float s = 0.f, q = 0.f;
  for (int r = r0; r < rend; ++r) {
    float v = h[(size_t)r * D + col];
    s += v; q += v * v;
  }
  atomicAdd(&sum[col], s);
  atomicAdd(&sq[col], q);
}

__global__ __launch_bounds__(128)
void bn_finalize_kernel(const float* __restrict__ sum, const float* __restrict__ sq,
                        const float* __restrict__ gamma, const float* __restrict__ beta,
                        float* __restrict__ scale, float* __restrict__ shift, int nN) {
  int c = threadIdx.x;
  float invn = 1.0f / (float)nN;
  float mean = sum[c] * invn;
  float var  = sq[c] * invn - mean * mean;   // biased variance (matches ref)
  float sc   = gamma[c] * rsqrtf(var + BN_EPS);
  scale[c] = sc;
  shift[c] = beta[c] - mean * sc;
}

__global__ __launch_bounds__(256)
void bn_apply_kernel(float* __restrict__ h, const float* __restrict__ scale,
                     const float* __restrict__ shift, int nQuads) {
  int i = blockIdx.x * blockDim.x + threadIdx.x;   // float4 index
  if (i >= nQuads) return;
  int c = (i & 31) * 4;                            // column base (D=128)
  float4 v = ((float4*)h)[i];
  v.x = v.x * scale[c + 0] + shift[c + 0];
  v.y = v.y * scale[c + 1] + shift[c + 1];
  v.z = v.z * scale[c + 2] + shift[c + 2];
  v.w = v.w * scale[c + 3] + shift[c + 3];
  ((float4*)h)[i] = v;
}

// --------------------- output SAGEConv (D->1) + sigmoid --------------------
// One wave per node: lanes reduce the 128-wide dot products.

__global__ __launch_bounds__(256)
void sage_out_kernel(const float* __restrict__ agg, const float* __restrict__ h,
                     const float* __restrict__ invdeg,
                     const float* __restrict__ Wl_out,
                     const float* __restrict__ Wr_out,
                     const float* __restrict__ b_out,
                     float* __restrict__ out, int nN) {
  int tid  = blockIdx.x * blockDim.x + threadIdx.x;
  int node = tid >> 5;
  int lane = tid & 31;
  if (node >= nN) return;
  float idg = invdeg[node];
  const float* ap = agg + (size_t)node * D;
  const float* hp = h   + (size_t)node * D;
  float s = 0.f;
#pragma unroll
  for (int c = 0; c < 4; ++c) {
    int k = lane + 32 * c;
    s += ap[k] * idg * Wl_out[k] + hp[k] * Wr_out[k];
  }
#pragma unroll
  for (int off = 16; off > 0; off >>= 1) s += __shfl_xor(s, off, 32);
  if (lane == 0) {
    float z = s + b_out[0];
    out[node] = 1.0f / (1.0f + __expf(-z));
  }
}

// ------------------------------- launcher ----------------------------------

extern "C" void kernel_launch(void* const* d_in, const int* in_sizes, int n_in,
                              void* d_out, int out_size, void* d_ws, size_t ws_size,
                              hipStream_t stream) {
  const float* x      = (const float*)d_in[0];
  const int*   ei     = (const int*)d_in[1];   // [2, E]: row0 = src, row1 = dst
  const float* Wl     = (const float*)d_in[2]; // [L, D, D]
  const float* Wr     = (const float*)d_in[3];
  const float* b      = (const float*)d_in[4]; // [L, D]
  const float* gamma  = (const float*)d_in[5];
  const float* beta   = (const float*)d_in[6];
  const float* Wl_out = (const float*)d_in[7]; // [D]
  const float* Wr_out = (const float*)d_in[8];
  const float* b_out  = (const float*)d_in[9];
  float*       out    = (float*)d_out;

  const int nN = in_sizes[0] / D;      // 100000
  const int nE = in_sizes[1] / 2;      // 1600000
  const int* src = ei;
  const int* dst = ei + nE;

  // workspace layout (floats)
  float* ws      = (float*)d_ws;
  float* deg     = ws;                       // N
  float* invdeg  = deg + nN;                 // N
  float* hA      = invdeg + nN;              // N*D
  float* hB      = hA + (size_t)nN * D;      // N*D
  float* agg     = hB + (size_t)nN * D;      // N*D
  float* bnsum   = agg + (size_t)nN * D;     // D
  float* bnsq    = bnsum + D;                // D
  float* bnscale = bnsq + D;                 // D
  float* bnshift = bnscale + D;              // D

  const int T = 256;
  const int nND  = nN * D;
  const int gFea = (nND + T - 1) / T;
  const int gN   = (nN + T - 1) / T;

  // degrees -> inverse degrees (recomputed every call: deterministic)
  fill_zero_kernel<<<gN, T, 0, stream>>>(deg, nN);
  deg_kernel<<<(nE + T - 1) / T, T, 0, stream>>>(dst, deg, nE, nN);
  invdeg_kernel<<<gN, T, 0, stream>>>(deg, invdeg, nN);

  const float* cur = x;      // layer input
  float* nxt = hA;
  float* other = hB;

  const int gScatter = ((nE * 32) + T - 1) / T;
  const int gGemm    = (nN + 15) / 16;
  const int gStats   = (nN + 255) / 256;
  const int gApply   = (nND / 4 + T - 1) / T;

  for (int l = 0; l < LAYERS; ++l) {
    // neighbor sum
    fill_zero_kernel<<<gFea, T, 0, stream>>>(agg, nND);
    scatter_kernel<<<gScatter, T, 0, stream>>>(src, dst, cur, agg, nE, nN);
    // fused (mean@Wl + h@Wr + b) with ReLU  -- WMMA fp32
    sage_gemm_relu_kernel<<<gGemm, T, 0, stream>>>(
        agg, cur, invdeg, Wl + (size_t)l * D * D, Wr + (size_t)l * D * D,
        b + (size_t)l * D, nxt, nN);
    // batchnorm
    fill_zero_kernel<<<1, T, 0, stream>>>(bnsum, 2 * D);  // bnsum & bnsq contiguous
    bn_stats_kernel<<<gStats, 128, 0, stream>>>(nxt, bnsum, bnsq, nN);
    bn_finalize_kernel<<<1, 128, 0, stream>>>(bnsum, bnsq, gamma + (size_t)l * D,
                                              beta + (size_t)l * D, bnscale,
                                              bnshift, nN);
    bn_apply_kernel<<<gApply, T, 0, stream>>>(nxt, bnscale, bnshift, nND / 4);
    // ping-pong
    cur = nxt;
    float* t = nxt; nxt = other; other = t;
    if (l == 0) { nxt = hB; other = hA; }   // after layer0: cur=hA, nxt=hB
  }

  // output layer: mean-aggregate final features, project to 1, sigmoid
  fill_zero_kernel<<<gFea, T, 0, stream>>>(agg, nND);
  scatter_kernel<<<gScatter, T, 0, stream>>>(src, dst, cur, agg, nE, nN);
  sage_out_kernel<<<((nN * 32) + T - 1) / T, T, 0, stream>>>(
      agg, cur, invdeg, Wl_out, Wr_out, b_out, out, nN);
}